// MMCL_INV_20813411516641
// MI455X (gfx1250) — compile-verified
//
#include <hip/hip_runtime.h>
#include <hip/hip_bf16.h>
#include <math.h>

// MMCL-style loss: normalize -> RBF kernel (WMMA f32 GEMM) -> 320 batched SPD
// solves recast as CG whose batched matvec is a single 320x320x320 fp32 GEMM
// (V_WMMA_F32_16X16X4_F32) plus rank-1 corrections -> scalar reductions.

#define BSZ   320
#define DIM   128
#define NZ    (2 * BSZ)          // 640
#define REG_F 0.1f
#define CG_ITERS 16

typedef __attribute__((ext_vector_type(2))) float v2f;
typedef __attribute__((ext_vector_type(8))) float v8f;

// ---------------------------------------------------------------- row normalize
__global__ void k_normalize(const float* __restrict__ z, float* __restrict__ zn) {
  const int row = blockIdx.x;       // 640 rows
  const int tid = threadIdx.x;      // 128 threads
  float v = z[row * DIM + tid];
  __shared__ float red[DIM];
  red[tid] = v * v;
  __syncthreads();
  for (int s = DIM / 2; s > 0; s >>= 1) {
    if (tid < s) red[tid] += red[tid + s];
    __syncthreads();
  }
  float nrm = sqrtf(red[0]);
  zn[row * DIM + tid] = v / nrm;
}

// ------------------------------------------------- K = exp(-g*(2-2*Z1@Z^T)) via WMMA
// One wave per 16x16 tile. A: 16x4 f32 (lane=M, K split by lane half, 2 VGPRs).
// B: 4x16 f32 (lane=N, K split by lane half). C: 8 VGPRs, M = v + 8*half, N = l16.
__global__ __launch_bounds__(32) void k_kernelmat(const float* __restrict__ zn,
                                                  float* __restrict__ Km) {
  const int tn   = blockIdx.x * 16;       // 0..639
  const int tm   = blockIdx.y * 16;       // 0..319
  const int lane = threadIdx.x;
  const int half = lane >> 4;
  const int l16  = lane & 15;
  const float* arow = zn + (size_t)(tm + l16) * DIM;
  const float* brow = zn + (size_t)(tn + l16) * DIM;   // B = Z^T: B[k][n] = zn[n][k]
  v8f acc = {};
  for (int k = 0; k < DIM; k += 4) {
    const int ka = k + 2 * half;
    v2f a, b;
    a.x = arow[ka];  a.y = arow[ka + 1];
    b.x = brow[ka];  b.y = brow[ka + 1];
    acc = __builtin_amdgcn_wmma_f32_16x16x4_f32(false, a, false, b,
                                                (short)0, acc, false, false);
  }
  const float gamma = 1.0f / 0.07f;
#pragma unroll
  for (int v = 0; v < 8; ++v) {
    const int m = tm + v + 8 * half;
    const int n = tn + l16;
    Km[(size_t)m * NZ + n] = expf(-gamma * (2.0f - 2.0f * acc[v]));
  }
}

// ------------------------------------------------- generic fp32 WMMA GEMM C = A@B
__global__ __launch_bounds__(32) void k_gemm_f32(const float* __restrict__ A, int lda,
                                                 const float* __restrict__ B, int ldb,
                                                 float* __restrict__ C, int ldc, int Kd) {
  const int tn   = blockIdx.x * 16;
  const int tm   = blockIdx.y * 16;
  const int lane = threadIdx.x;
  const int half = lane >> 4;
  const int l16  = lane & 15;
  v8f acc = {};
  for (int k = 0; k < Kd; k += 4) {
    const int ka = k + 2 * half;
    v2f a, b;
    a.x = A[(size_t)(tm + l16) * lda + ka];
    a.y = A[(size_t)(tm + l16) * lda + ka + 1];
    b.x = B[(size_t)ka * ldb + tn + l16];
    b.y = B[(size_t)(ka + 1) * ldb + tn + l16];
    acc = __builtin_amdgcn_wmma_f32_16x16x4_f32(false, a, false, b,
                                                (short)0, acc, false, false);
  }
#pragma unroll
  for (int v = 0; v < 8; ++v)
    C[(size_t)(tm + v + 8 * half) * ldc + tn + l16] = acc[v];
}

// ---------------------------------------------------------------- block reduce(320)
__device__ __forceinline__ float block_reduce_320(float v, float* red, int i) {
  red[i] = v;
  __syncthreads();
  if (i < 64) red[i] += red[i + 256];
  __syncthreads();
  for (int s = 128; s > 0; s >>= 1) {
    if (i < s) red[i] += red[i + s];
    __syncthreads();
  }
  float r = red[0];
  __syncthreads();
  return r;
}

// ---------------------------------------------------------------- CG init
__global__ void k_cg_init(float* __restrict__ X, float* __restrict__ R,
                          float* __restrict__ P, float* __restrict__ rs) {
  const int t = blockIdx.x;
  const int i = threadIdx.x;
  const float r = (i == t) ? 0.0f : 2.0f;   // rhs = 2, coordinate t pinned to 0
  X[t * BSZ + i] = 0.0f;
  R[t * BSZ + i] = r;
  P[t * BSZ + i] = r;
  if (i == 0) rs[t] = 4.0f * (float)(BSZ - 1);
}

// --------------------------------------- CG step (after Q0 = P @ KK GEMM)
// (DD_t p)_i = (KK p)_i + s*(1 - KK[t][i]) + 0.1*p_i - (KK p)_t , with p_t = 0.
__global__ void k_cg_update(const float* __restrict__ Km, const float* __restrict__ Q0,
                            float* __restrict__ X, float* __restrict__ R,
                            float* __restrict__ P, float* __restrict__ rs) {
  __shared__ float red[BSZ];
  const int t = blockIdx.x;
  const int i = threadIdx.x;
  const float rs_old = rs[t];
  const float p = P[t * BSZ + i];
  const float s = block_reduce_320(p, red, i);
  const float d = Q0[t * BSZ + t];
  float q = Q0[t * BSZ + i] + s * (1.0f - Km[(size_t)t * NZ + i]) + REG_F * p - d;
  if (i == t) q = 0.0f;
  const float pq = block_reduce_320(p * q, red, i);
  const float alpha = (pq != 0.0f) ? rs_old / pq : 0.0f;
  const float x = X[t * BSZ + i] + alpha * p;
  const float r = R[t * BSZ + i] - alpha * q;
  X[t * BSZ + i] = x;
  R[t * BSZ + i] = r;
  const float rs_new = block_reduce_320(r * r, red, i);
  const float beta = (rs_old != 0.0f) ? rs_new / rs_old : 0.0f;
  P[t * BSZ + i] = r + beta * p;
  if (i == 0) rs[t] = rs_new;
}

// ---------------------------------------------------------------- per-t partials
__global__ void k_finalize(const float* __restrict__ Km, const float* __restrict__ X,
                           float* __restrict__ part) {
  __shared__ float red[BSZ];
  const int t = blockIdx.x;
  const int i = threadIdx.x;
  const float x = X[t * BSZ + i];
  const float c = fminf(fmaxf(x, 0.0f), 1.0f);     // clip(relu(x), 0, C_BOX)
  const bool excl = (i == t);
  const float ks = Km[(size_t)i * NZ + BSZ + t];    // Ks[i, t] = K[i][320+t]
  const float ax  = block_reduce_320(excl ? 0.0f : c,        red, i);
  const float neg = block_reduce_320(excl ? 0.0f : c * ks,   red, i);
  const float kn  = block_reduce_320(excl ? 0.0f : ks,       red, i);
  const float cz  = block_reduce_320((!excl && c == 0.0f) ? 1.0f : 0.0f, red, i);
  const float co  = block_reduce_320((!excl && c == 1.0f) ? 1.0f : 0.0f, red, i);
  const float cp  = block_reduce_320((!excl && c >  0.0f) ? 1.0f : 0.0f, red, i);
  if (i == 0) {
    const float pt = Km[(size_t)t * NZ + BSZ + t];  // diag(Ks)
    part[0 * BSZ + t] = ax * pt;
    part[1 * BSZ + t] = neg;
    part[2 * BSZ + t] = pt;
    part[3 * BSZ + t] = kn;
    part[4 * BSZ + t] = cz;
    part[5 * BSZ + t] = co;
    part[6 * BSZ + t] = cp;
  }
}

// ---------------------------------------------------------------- final scalars
__global__ void k_output(const float* __restrict__ part, float* __restrict__ out) {
  __shared__ float red[BSZ];
  const int i = threadIdx.x;
  const float s0 = block_reduce_320(part[0 * BSZ + i], red, i);
  const float s1 = block_reduce_320(part[1 * BSZ + i], red, i);
  const float s2 = block_reduce_320(part[2 * BSZ + i], red, i);
  const float s3 = block_reduce_320(part[3 * BSZ + i], red, i);
  const float s4 = block_reduce_320(part[4 * BSZ + i], red, i);
  const float s5 = block_reduce_320(part[5 * BSZ + i], red, i);
  const float s6 = block_reduce_320(part[6 * BSZ + i], red, i);
  if (i == 0) {
    const float pos_loss = s0 / (float)BSZ;
    const float neg_loss = s1 / (float)BSZ;
    out[0] = neg_loss - pos_loss;                        // loss
    out[1] = s2 / (float)BSZ;                            // pos_terms.mean()
    out[2] = s3 / (float)(BSZ * (BSZ - 1));              // Kn.mean()
    out[3] = s5 / (s6 + 1e-10f);                         // sparsity
    out[4] = s4 / (float)(BSZ * (BSZ - 1));              // num_zero
    out[5] = 0.0f;
  }
}

extern "C" void kernel_launch(void* const* d_in, const int* in_sizes, int n_in,
                              void* d_out, int out_size, void* d_ws, size_t ws_size,
                              hipStream_t stream) {
  (void)in_sizes; (void)n_in; (void)out_size; (void)ws_size;
  const float* z = (const float*)d_in[0];     // (640, 128) fp32
  float* out = (float*)d_out;                 // 6 fp32 scalars

  float* w    = (float*)d_ws;
  float* zn   = w;                    // 640*128
  float* Km   = zn + NZ * DIM;        // 320*640
  float* X    = Km + BSZ * NZ;        // 320*320
  float* R    = X  + BSZ * BSZ;
  float* P    = R  + BSZ * BSZ;
  float* Q0   = P  + BSZ * BSZ;
  float* rs   = Q0 + BSZ * BSZ;       // 320
  float* part = rs + BSZ;             // 7*320

  k_normalize<<<NZ, DIM, 0, stream>>>(z, zn);
  k_kernelmat<<<dim3(NZ / 16, BSZ / 16), 32, 0, stream>>>(zn, Km);
  k_cg_init<<<BSZ, BSZ, 0, stream>>>(X, R, P, rs);
  for (int it = 0; it < CG_ITERS; ++it) {
    // Q0 = P (320x320) @ KK (KK = left 320 cols of Km, row-major ldb=640)
    k_gemm_f32<<<dim3(BSZ / 16, BSZ / 16), 32, 0, stream>>>(P, BSZ, Km, NZ, Q0, BSZ, BSZ);
    k_cg_update<<<BSZ, BSZ, 0, stream>>>(Km, Q0, X, R, P, rs);
  }
  k_finalize<<<BSZ, BSZ, 0, stream>>>(Km, X, part);
  k_output<<<1, BSZ, 0, stream>>>(part, out);
}